// Net_15650860827522
// MI455X (gfx1250) — compile-verified
//
#include <hip/hip_runtime.h>

typedef __attribute__((ext_vector_type(2))) float v2f;
typedef __attribute__((ext_vector_type(8))) float v8f;

// ---------------------------------------------------------------------------
// Phase 1: in-degree (edges only; +1 self loop added in k_dinv)
// ---------------------------------------------------------------------------
__global__ __launch_bounds__(256) void k_degree(const int* __restrict__ edst,
                                                unsigned* __restrict__ deg,
                                                int E) {
    int e = blockIdx.x * blockDim.x + threadIdx.x;
    if (e < E) atomicAdd(&deg[edst[e]], 1u);
}

// deg (u32) -> dinv (f32), in place. +1 for the self loop.
__global__ __launch_bounds__(256) void k_dinv(float* degdinv, int n) {
    int i = blockIdx.x * blockDim.x + threadIdx.x;
    if (i < n) {
        unsigned d = ((const unsigned*)degdinv)[i];
        degdinv[i] = rsqrtf((float)(d + 1u));
    }
}

// ---------------------------------------------------------------------------
// Phase 2 (layer 1): u = (x @ W1) * dinv  via V_WMMA_F32_16X16X4_F32.
// One wave handles 16 nodes. K = 34 padded to 36 (9 chunks of 4).
// B-chunk vectors are wave-invariant: preloaded branchlessly (clamped index +
// cndmask select, no exec-mask branches), hoisted out of the K loop.
// Also writes s = u (self-loop initialization of the aggregation buffer).
// ---------------------------------------------------------------------------
__global__ __launch_bounds__(256) void k_l1_wmma(const float* __restrict__ x,
                                                 const float* __restrict__ W1,
                                                 const float* __restrict__ dinv,
                                                 float* __restrict__ u,
                                                 float* __restrict__ s,
                                                 int n) {
    int wave = (blockIdx.x * blockDim.x + threadIdx.x) >> 5;
    int lane = threadIdx.x & 31;
    int base = wave * 16;
    if (base >= n) return;                 // wave-uniform: EXEC stays all-ones

    int row  = base + (lane & 15);
    int col  = lane & 15;
    int khi  = (lane >> 4) * 2;            // 0 for lanes 0-15, 2 for 16-31
    const float* xr = x + (size_t)row * 34;
    bool cvalid = (col < 4);

    // Preload all 9 B chunks (4x16 slices of W1, zero-padded), branchless.
    v2f bv[9];
#pragma unroll
    for (int c = 0; c < 9; ++c) {
        int k0 = c * 4 + khi;
        int k1 = k0 + 1;
        float w0 = W1[min(k0 * 4 + col, 135)];   // unconditional, clamped
        float w1 = W1[min(k1 * 4 + col, 135)];
        bv[c].x = (cvalid && k0 < 34) ? w0 : 0.0f;
        bv[c].y = (cvalid && k1 < 34) ? w1 : 0.0f;
    }

    v8f acc = {};
#pragma unroll
    for (int c = 0; c < 9; ++c) {
        int ka0 = c * 4 + khi;
        int ka1 = ka0 + 1;
        v2f a;
        float a0 = xr[min(ka0, 33)];             // unconditional, clamped
        float a1 = xr[min(ka1, 33)];
        a.x = (ka0 < 34) ? a0 : 0.0f;
        a.y = (ka1 < 34) ? a1 : 0.0f;
        acc = __builtin_amdgcn_wmma_f32_16x16x4_f32(
            /*neg_a=*/false, a, /*neg_b=*/false, bv[c],
            /*c_mod=*/(short)0, acc, /*reuse_a=*/false, /*reuse_b=*/false);
    }

    if (cvalid) {                          // only output columns 0..3 are real
        int mbase = base + ((lane >> 4) << 3);
#pragma unroll
        for (int v = 0; v < 8; ++v) {
            int r = mbase + v;
            float val = acc[v] * dinv[r];
            u[(size_t)r * 4 + col] = val;
            s[(size_t)r * 4 + col] = val;
        }
    }
}

// ---------------------------------------------------------------------------
// Edge scatter: s[dst] += u[src]  (4-wide and 2-wide variants)
// ---------------------------------------------------------------------------
__global__ __launch_bounds__(256) void k_edge4(const int* __restrict__ esrc,
                                               const int* __restrict__ edst,
                                               const float* __restrict__ u,
                                               float* __restrict__ s, int E) {
    int e = blockIdx.x * blockDim.x + threadIdx.x;
    if (e >= E) return;
    int a = esrc[e], d = edst[e];
    float4 uv = ((const float4*)u)[a];
    float* sp = s + (size_t)d * 4;
    atomicAdd(sp + 0, uv.x);
    atomicAdd(sp + 1, uv.y);
    atomicAdd(sp + 2, uv.z);
    atomicAdd(sp + 3, uv.w);
}

__global__ __launch_bounds__(256) void k_edge2(const int* __restrict__ esrc,
                                               const int* __restrict__ edst,
                                               const float* __restrict__ u2,
                                               float* __restrict__ s2, int E) {
    int e = blockIdx.x * blockDim.x + threadIdx.x;
    if (e >= E) return;
    int a = esrc[e], d = edst[e];
    float2 uv = ((const float2*)u2)[a];
    float* sp = s2 + (size_t)d * 2;
    atomicAdd(sp + 0, uv.x);
    atomicAdd(sp + 1, uv.y);
}

// ---------------------------------------------------------------------------
// Fused node kernels: h = relu(s*dinv + b); u_next = (h @ W) * dinv; s_next=u
// ---------------------------------------------------------------------------
__global__ __launch_bounds__(256) void k_node_4to4(const float* __restrict__ b,
                                                   const float* __restrict__ W,
                                                   const float* __restrict__ dinv,
                                                   float* s,        // in s1 / out s2 (aliased ok)
                                                   float* u,        // out u2
                                                   int n) {
    int i = blockIdx.x * blockDim.x + threadIdx.x;
    if (i >= n) return;
    float di = dinv[i];
    float4 sv = ((const float4*)s)[i];
    float h0 = fmaxf(fmaf(sv.x, di, b[0]), 0.0f);
    float h1 = fmaxf(fmaf(sv.y, di, b[1]), 0.0f);
    float h2 = fmaxf(fmaf(sv.z, di, b[2]), 0.0f);
    float h3 = fmaxf(fmaf(sv.w, di, b[3]), 0.0f);
    float4 o;
    o.x = (h0 * W[0] + h1 * W[4] + h2 * W[8]  + h3 * W[12]) * di;
    o.y = (h0 * W[1] + h1 * W[5] + h2 * W[9]  + h3 * W[13]) * di;
    o.z = (h0 * W[2] + h1 * W[6] + h2 * W[10] + h3 * W[14]) * di;
    o.w = (h0 * W[3] + h1 * W[7] + h2 * W[11] + h3 * W[15]) * di;
    ((float4*)u)[i] = o;
    ((float4*)s)[i] = o;
}

__global__ __launch_bounds__(256) void k_node_4to2(const float* __restrict__ b,
                                                   const float* __restrict__ W,  // 4x2
                                                   const float* __restrict__ dinv,
                                                   const float* __restrict__ s,  // in s2 (stride 4)
                                                   float* __restrict__ u2,       // out u3 (stride 2)
                                                   float* __restrict__ s2,       // out s3 (stride 2)
                                                   int n) {
    int i = blockIdx.x * blockDim.x + threadIdx.x;
    if (i >= n) return;
    float di = dinv[i];
    float4 sv = ((const float4*)s)[i];
    float h0 = fmaxf(fmaf(sv.x, di, b[0]), 0.0f);
    float h1 = fmaxf(fmaf(sv.y, di, b[1]), 0.0f);
    float h2 = fmaxf(fmaf(sv.z, di, b[2]), 0.0f);
    float h3 = fmaxf(fmaf(sv.w, di, b[3]), 0.0f);
    float2 o;
    o.x = (h0 * W[0] + h1 * W[2] + h2 * W[4] + h3 * W[6]) * di;
    o.y = (h0 * W[1] + h1 * W[3] + h2 * W[5] + h3 * W[7]) * di;
    ((float2*)u2)[i] = o;
    ((float2*)s2)[i] = o;
}

// h3 = relu(s3*dinv + b3); out = h3 @ Wl + bl.  Writes both outputs.
__global__ __launch_bounds__(256) void k_out(const float* __restrict__ b3,
                                             const float* __restrict__ Wl,  // 2x4
                                             const float* __restrict__ bl,
                                             const float* __restrict__ dinv,
                                             const float* __restrict__ s3,  // stride 2
                                             float* __restrict__ out,       // [n,4]
                                             float* __restrict__ hout,      // [n,2]
                                             int n) {
    int i = blockIdx.x * blockDim.x + threadIdx.x;
    if (i >= n) return;
    float di = dinv[i];
    float2 sv = ((const float2*)s3)[i];
    float h0 = fmaxf(fmaf(sv.x, di, b3[0]), 0.0f);
    float h1 = fmaxf(fmaf(sv.y, di, b3[1]), 0.0f);
    float2 hv; hv.x = h0; hv.y = h1;
    ((float2*)hout)[i] = hv;
    float4 o;
    o.x = fmaf(h0, Wl[0], fmaf(h1, Wl[4], bl[0]));
    o.y = fmaf(h0, Wl[1], fmaf(h1, Wl[5], bl[1]));
    o.z = fmaf(h0, Wl[2], fmaf(h1, Wl[6], bl[2]));
    o.w = fmaf(h0, Wl[3], fmaf(h1, Wl[7], bl[3]));
    ((float4*)out)[i] = o;
}

// ---------------------------------------------------------------------------
extern "C" void kernel_launch(void* const* d_in, const int* in_sizes, int n_in,
                              void* d_out, int out_size, void* d_ws, size_t ws_size,
                              hipStream_t stream) {
    const float* x  = (const float*)d_in[0];
    const int*   ei = (const int*)d_in[1];
    const float* W1 = (const float*)d_in[2];
    const float* b1 = (const float*)d_in[3];
    const float* W2 = (const float*)d_in[4];
    const float* b2 = (const float*)d_in[5];
    const float* W3 = (const float*)d_in[6];
    const float* b3 = (const float*)d_in[7];
    const float* Wl = (const float*)d_in[8];
    const float* bl = (const float*)d_in[9];

    const int n = in_sizes[0] / 34;
    const int E = in_sizes[1] / 2;
    const int* esrc = ei;
    const int* edst = ei + E;

    // Workspace layout (256B-aligned regions)
    char*  ws   = (char*)d_ws;
    size_t offA = 0;                                             // dinv/deg : n*4 B
    size_t offB = (offA + (size_t)n * 4  + 255) & ~(size_t)255;  // u : n*16 B
    size_t offC = (offB + (size_t)n * 16 + 255) & ~(size_t)255;  // s : n*16 B
    float* dinv = (float*)(ws + offA);
    float* u    = (float*)(ws + offB);
    float* sbuf = (float*)(ws + offC);
    float* u3   = u;                        // n*2 floats (u buffer retired)
    float* s3   = u + (size_t)n * 2;        // n*2 floats

    float* out  = (float*)d_out;            // [n,4]
    float* hout = out + (size_t)n * 4;      // [n,2]

    const int B  = 256;
    const int gE = (E + B - 1) / B;
    const int gN = (n + B - 1) / B;
    const int waves = (n + 15) / 16;
    const int gW = (waves * 32 + B - 1) / B;

    // degrees -> dinv
    hipMemsetAsync(dinv, 0, (size_t)n * 4, stream);
    k_degree<<<gE, B, 0, stream>>>(edst, (unsigned*)dinv, E);
    k_dinv<<<gN, B, 0, stream>>>(dinv, n);

    // layer 1: WMMA transform (+ self-loop init), then edge scatter
    k_l1_wmma<<<gW, B, 0, stream>>>(x, W1, dinv, u, sbuf, n);
    k_edge4<<<gE, B, 0, stream>>>(esrc, edst, u, sbuf, E);

    // layer 2
    k_node_4to4<<<gN, B, 0, stream>>>(b1, W2, dinv, sbuf, u, n);
    k_edge4<<<gE, B, 0, stream>>>(esrc, edst, u, sbuf, E);

    // layer 3 (width 2)
    k_node_4to2<<<gN, B, 0, stream>>>(b2, W3, dinv, sbuf, u3, s3, n);
    k_edge2<<<gE, B, 0, stream>>>(esrc, edst, u3, s3, E);

    // outputs
    k_out<<<gN, B, 0, stream>>>(b3, Wl, bl, dinv, s3, out, hout, n);
}